// HybridModel_33036888441062
// MI455X (gfx1250) — compile-verified
//
#include <hip/hip_runtime.h>
#include <math.h>

typedef __attribute__((ext_vector_type(2))) float v2f;
typedef __attribute__((ext_vector_type(8))) float v8f;

#define BATCH    16384
#define N_IN     64
#define H1       32
#define NQ       10
#define NL       6
#define PI_F     3.14159265358979323846f

// ---------------------------------------------------------------------------
// Kernel 1: h = relu(x @ W1^T + b1) via V_WMMA_F32_16X16X4_F32
// One wave computes a 16(M) x 32(N) tile; K=64 in 16 steps of 4.
// A (16x4 f32): lanes 0-15 hold M=0..15,K={0,1}; lanes 16-31 hold K={2,3}.
// B (4x16 f32): assumed mirrored K layout, N = lane&15.
// C (16x16 f32): VGPR v -> M=v (lanes 0-15) / M=v+8 (lanes 16-31), N=lane&15.
// ---------------------------------------------------------------------------
__global__ void mlp1_wmma(const float* __restrict__ x,
                          const float* __restrict__ W1,
                          const float* __restrict__ b1,
                          float* __restrict__ h)
{
    int wave = (blockIdx.x * blockDim.x + threadIdx.x) >> 5;   // 0..1023
    int lane = threadIdx.x & 31;
    int m0   = wave * 16;
    int mn   = lane & 15;            // row-in-tile for A, col-in-tile for B
    int ko   = (lane >> 4) * 2;      // K sub-offset per half-wave

    v8f acc0 = {};                   // N tile 0..15
    v8f acc1 = {};                   // N tile 16..31

    const float* xrow  = x  + (size_t)(m0 + mn) * N_IN;
    const float* w1r0  = W1 + (size_t)(mn)      * N_IN;   // output col mn
    const float* w1r1  = W1 + (size_t)(16 + mn) * N_IN;   // output col 16+mn

    #pragma unroll
    for (int kk = 0; kk < N_IN; kk += 4) {
        v2f a;  a.x  = xrow[kk + ko];  a.y  = xrow[kk + ko + 1];
        v2f bA; bA.x = w1r0[kk + ko];  bA.y = w1r0[kk + ko + 1];
        v2f bB; bB.x = w1r1[kk + ko];  bB.y = w1r1[kk + ko + 1];
        acc0 = __builtin_amdgcn_wmma_f32_16x16x4_f32(false, a, false, bA,
                                                     (short)0, acc0, false, false);
        acc1 = __builtin_amdgcn_wmma_f32_16x16x4_f32(false, a, false, bB,
                                                     (short)0, acc1, false, false);
    }

    int rowHalf = (lane >> 4) * 8;
    float bias0 = b1[mn];
    float bias1 = b1[16 + mn];
    #pragma unroll
    for (int v = 0; v < 8; ++v) {
        int row = m0 + rowHalf + v;
        float c0 = acc0[v] + bias0;  c0 = c0 > 0.f ? c0 : 0.f;
        float c1 = acc1[v] + bias1;  c1 = c1 > 0.f ? c1 : 0.f;
        h[(size_t)row * H1 + mn]      = c0;
        h[(size_t)row * H1 + 16 + mn] = c1;
    }
}

// ---------------------------------------------------------------------------
// Kernel 2: fuse RZ(tz)*RY(ty)*RX(tx) into one 2x2 complex matrix per
// (layer, qubit). 60 matrices x 8 floats.
// ---------------------------------------------------------------------------
struct cpx { float r, i; };
__device__ __forceinline__ cpx cmul(cpx a, cpx b) {
    return { a.r*b.r - a.i*b.i, a.r*b.i + a.i*b.r };
}

__global__ void prep_gates(const float* __restrict__ qp, float* __restrict__ U)
{
    int t = blockIdx.x * blockDim.x + threadIdx.x;
    if (t >= NL * NQ) return;
    const float* w = qp + t * 3;
    float tx = w[0] * 0.5f, ty = w[1] * 0.5f, tz = w[2] * 0.5f;
    float cx = cosf(tx), sx = sinf(tx);
    float cy = cosf(ty), sy = sinf(ty);
    float cz = cosf(tz), sz = sinf(tz);
    // A = RY * RX
    cpx a00 = { cy*cx,  sy*sx };
    cpx a01 = { -sy*cx, -cy*sx };
    cpx a10 = { sy*cx,  -cy*sx };
    cpx a11 = { cy*cx,  -sy*sx };
    // U = RZ * A ;  RZ = diag(e^{-i tz}, e^{+i tz})
    cpx e0 = { cz, -sz }, e1 = { cz, sz };
    cpx u00 = cmul(e0, a00), u01 = cmul(e0, a01);
    cpx u10 = cmul(e1, a10), u11 = cmul(e1, a11);
    float* o = U + t * 8;
    o[0]=u00.r; o[1]=u00.i; o[2]=u01.r; o[3]=u01.i;
    o[4]=u10.r; o[5]=u10.i; o[6]=u11.r; o[7]=u11.i;
}

// ---------------------------------------------------------------------------
// Kernel 3: state-vector simulator, one wave32 per sample.
// Amplitude index: qubits 0..4 -> register bits 0..4 (32 complex regs/lane),
// qubits 5..9 -> lane bits 0..4.
// ---------------------------------------------------------------------------
__device__ __forceinline__ float wsum(float v) {
    v += __shfl_xor(v, 1, 32);  v += __shfl_xor(v, 2, 32);
    v += __shfl_xor(v, 4, 32);  v += __shfl_xor(v, 8, 32);
    v += __shfl_xor(v, 16, 32);
    return v;
}

template<int B>
__device__ __forceinline__ void gate_reg(float* sr, float* si,
    float m00r,float m00i,float m01r,float m01i,
    float m10r,float m10i,float m11r,float m11i)
{
    #pragma unroll
    for (int r = 0; r < 32; ++r) {
        if (!(r & (1 << B))) {
            int r1 = r | (1 << B);
            float a0r = sr[r],  a0i = si[r];
            float a1r = sr[r1], a1i = si[r1];
            sr[r]  = m00r*a0r - m00i*a0i + m01r*a1r - m01i*a1i;
            si[r]  = m00r*a0i + m00i*a0r + m01r*a1i + m01i*a1r;
            sr[r1] = m10r*a0r - m10i*a0i + m11r*a1r - m11i*a1i;
            si[r1] = m10r*a0i + m10i*a0r + m11r*a1i + m11i*a1r;
        }
    }
}

template<int B>
__device__ __forceinline__ void gate_lane(float* sr, float* si, int lane,
    float m00r,float m00i,float m01r,float m01i,
    float m10r,float m10i,float m11r,float m11i)
{
    bool hi = (lane >> B) & 1;
    float c0r = hi ? m10r : m00r,  c0i = hi ? m10i : m00i;
    float c1r = hi ? m11r : m01r,  c1i = hi ? m11i : m01i;
    #pragma unroll
    for (int r = 0; r < 32; ++r) {
        float pr  = __shfl_xor(sr[r], 1 << B, 32);
        float pim = __shfl_xor(si[r], 1 << B, 32);
        float a0r = hi ? pr  : sr[r],  a0i = hi ? pim : si[r];
        float a1r = hi ? sr[r] : pr,   a1i = hi ? si[r] : pim;
        sr[r] = c0r*a0r - c0i*a0i + c1r*a1r - c1i*a1i;
        si[r] = c0r*a0i + c0i*a0r + c1r*a1i + c1i*a1r;
    }
}

template<int C>  // control reg-bit C, target reg-bit C+1 (C = 0..3)
__device__ __forceinline__ void cnot_rr(float* sr, float* si)
{
    #pragma unroll
    for (int r = 0; r < 32; ++r) {
        if (((r >> C) & 1) && !((r >> (C + 1)) & 1)) {
            int r2 = r | (1 << (C + 1));
            float t;
            t = sr[r]; sr[r] = sr[r2]; sr[r2] = t;
            t = si[r]; si[r] = si[r2]; si[r2] = t;
        }
    }
}

// control = reg bit 4 (qubit 4), target = lane bit 0 (qubit 5)
__device__ __forceinline__ void cnot_c4(float* sr, float* si)
{
    #pragma unroll
    for (int r = 16; r < 32; ++r) {
        sr[r] = __shfl_xor(sr[r], 1, 32);
        si[r] = __shfl_xor(si[r], 1, 32);
    }
}

template<int CB>  // control lane-bit CB, target lane-bit CB+1 (CB = 0..3)
__device__ __forceinline__ void cnot_ll(float* sr, float* si, int lane)
{
    int src = lane ^ ((((lane >> CB) & 1)) << (CB + 1));
    #pragma unroll
    for (int r = 0; r < 32; ++r) {
        sr[r] = __shfl(sr[r], src, 32);
        si[r] = __shfl(si[r], src, 32);
    }
}

__global__ void __launch_bounds__(256)
qsim_kernel(const float* __restrict__ h,
            const float* __restrict__ W2, const float* __restrict__ b2,
            const float* __restrict__ Uws,
            const float* __restrict__ W3, const float* __restrict__ b3,
            const float* __restrict__ W4, const float* __restrict__ b4,
            float* __restrict__ out)
{
    int s    = (blockIdx.x * blockDim.x + threadIdx.x) >> 5;   // sample
    int lane = threadIdx.x & 31;

    // ---- angles = tanh(h @ W2^T + b2) * pi ----
    float hv = h[(size_t)s * H1 + lane];
    float ang[NQ];
    #pragma unroll
    for (int i = 0; i < NQ; ++i) {
        float p = W2[i * H1 + lane] * hv;
        p = wsum(p);
        ang[i] = tanhf(p + b2[i]) * PI_F;
    }

    // ---- init |0...0> ----
    float sr[32], si[32];
    #pragma unroll
    for (int r = 0; r < 32; ++r) { sr[r] = 0.f; si[r] = 0.f; }
    sr[0] = (lane == 0) ? 1.0f : 0.0f;

    // ---- encoding RY(angle_i) on qubit i ----
    #define ENC_REG(Q) { float th = ang[Q]*0.5f, c = cosf(th), sn = sinf(th); \
        gate_reg<Q>(sr, si, c,0.f, -sn,0.f, sn,0.f, c,0.f); }
    #define ENC_LANE(Q) { float th = ang[Q]*0.5f, c = cosf(th), sn = sinf(th); \
        gate_lane<(Q)-5>(sr, si, lane, c,0.f, -sn,0.f, sn,0.f, c,0.f); }
    ENC_REG(0) ENC_REG(1) ENC_REG(2) ENC_REG(3) ENC_REG(4)
    ENC_LANE(5) ENC_LANE(6) ENC_LANE(7) ENC_LANE(8) ENC_LANE(9)

    // ---- 6 layers: fused RZ*RY*RX per qubit, then CNOT chain ----
    #pragma unroll 1
    for (int l = 0; l < NL; ++l) {
        const float* Ub = Uws + l * (NQ * 8);
        #define FG_REG(Q)  { const float* u = Ub + (Q)*8; \
            gate_reg<Q>(sr, si, u[0],u[1],u[2],u[3],u[4],u[5],u[6],u[7]); }
        #define FG_LANE(Q) { const float* u = Ub + (Q)*8; \
            gate_lane<(Q)-5>(sr, si, lane, u[0],u[1],u[2],u[3],u[4],u[5],u[6],u[7]); }
        FG_REG(0) FG_REG(1) FG_REG(2) FG_REG(3) FG_REG(4)
        FG_LANE(5) FG_LANE(6) FG_LANE(7) FG_LANE(8) FG_LANE(9)

        cnot_rr<0>(sr, si);
        cnot_rr<1>(sr, si);
        cnot_rr<2>(sr, si);
        cnot_rr<3>(sr, si);
        cnot_c4(sr, si);
        cnot_ll<0>(sr, si, lane);
        cnot_ll<1>(sr, si, lane);
        cnot_ll<2>(sr, si, lane);
        cnot_ll<3>(sr, si, lane);
    }

    // ---- measurement: z_q = sum (+/-) |amp|^2 ----
    float P = 0.f, T0 = 0.f, T1 = 0.f, T2 = 0.f, T3 = 0.f, T4 = 0.f;
    #pragma unroll
    for (int r = 0; r < 32; ++r) {
        float p = sr[r]*sr[r] + si[r]*si[r];
        P  += p;
        T0 += ((r >> 0) & 1) ? -p : p;
        T1 += ((r >> 1) & 1) ? -p : p;
        T2 += ((r >> 2) & 1) ? -p : p;
        T3 += ((r >> 3) & 1) ? -p : p;
        T4 += ((r >> 4) & 1) ? -p : p;
    }
    float z[NQ];
    z[0] = wsum(T0); z[1] = wsum(T1); z[2] = wsum(T2);
    z[3] = wsum(T3); z[4] = wsum(T4);
    #pragma unroll
    for (int j = 0; j < 5; ++j)
        z[5 + j] = wsum(((lane >> j) & 1) ? -P : P);

    // ---- tail MLP: h2 = relu(z @ W3^T + b3); out = h2 @ W4^T + b4 ----
    float h2[16];
    #pragma unroll
    for (int i = 0; i < 16; ++i) {
        float a = b3[i];
        #pragma unroll
        for (int q = 0; q < NQ; ++q) a += W3[i * NQ + q] * z[q];
        h2[i] = a > 0.f ? a : 0.f;
    }
    float o0 = b4[0], o1 = b4[1];
    #pragma unroll
    for (int i = 0; i < 16; ++i) {
        o0 += W4[i]      * h2[i];
        o1 += W4[16 + i] * h2[i];
    }
    if (lane == 0) {
        out[(size_t)s * 2 + 0] = o0;
        out[(size_t)s * 2 + 1] = o1;
    }
}

// ---------------------------------------------------------------------------
extern "C" void kernel_launch(void* const* d_in, const int* in_sizes, int n_in,
                              void* d_out, int out_size, void* d_ws, size_t ws_size,
                              hipStream_t stream)
{
    const float* x  = (const float*)d_in[0];
    const float* W1 = (const float*)d_in[1];
    const float* b1 = (const float*)d_in[2];
    const float* W2 = (const float*)d_in[3];
    const float* b2 = (const float*)d_in[4];
    const float* qp = (const float*)d_in[5];
    const float* W3 = (const float*)d_in[6];
    const float* b3 = (const float*)d_in[7];
    const float* W4 = (const float*)d_in[8];
    const float* b4 = (const float*)d_in[9];
    float* out = (float*)d_out;

    float* Uws = (float*)d_ws;                          // 60*8 floats
    float* hws = (float*)((char*)d_ws + 4096);          // 16384*32 floats (2 MB)

    prep_gates<<<1, 64, 0, stream>>>(qp, Uws);

    // 1024 waves (16 rows each) -> 128 blocks of 256 threads
    mlp1_wmma<<<BATCH / 16 / 8, 256, 0, stream>>>(x, W1, b1, hws);

    // 16384 waves -> 2048 blocks of 256 threads
    qsim_kernel<<<BATCH / 8, 256, 0, stream>>>(hws, W2, b2, Uws,
                                               W3, b3, W4, b4, out);
}